// Simple_GNN_1494648619090
// MI455X (gfx1250) — compile-verified
//
#include <hip/hip_runtime.h>

typedef float v2f __attribute__((ext_vector_type(2)));
typedef float v8f __attribute__((ext_vector_type(8)));

#define IN_F 64
#define INT_F 128
#define OUT_F 32
#define GNN_DEPTH 4
#define NEG_SLOPE 0.01f

__device__ __forceinline__ void atomic_add_f32(float* p, float v) {
  __hip_atomic_fetch_add(p, v, __ATOMIC_RELAXED, __HIP_MEMORY_SCOPE_AGENT);
}

// ---- degree counting -------------------------------------------------------
__global__ void count_edges_kernel(const int* __restrict__ edge_index,
                                   float* __restrict__ cnt, int n_edges) {
  int e = blockIdx.x * blockDim.x + threadIdx.x;
  if (e < n_edges) {
    int d = edge_index[e * 2 + 1];
    atomic_add_f32(&cnt[d], 1.0f);
  }
}

__global__ void finalize_rdenom_kernel(float* __restrict__ cnt, int n_nodes) {
  int i = blockIdx.x * blockDim.x + threadIdx.x;
  if (i < n_nodes) cnt[i] = 1.0f / fmaxf(cnt[i], 1.0f);
}

// ---- edge scatter: agg[dst] += w * h[src] ---------------------------------
// one thread per (edge, 4-float chunk): 32 chunks cover INT_F=128 features
__global__ void scatter_kernel(const float* __restrict__ h,
                               const int* __restrict__ edge_index,
                               const float* __restrict__ ew,
                               float* __restrict__ agg, int n_edges) {
  int gid = blockIdx.x * blockDim.x + threadIdx.x;
  int e = gid >> 5;
  if (e >= n_edges) return;
  int c = (gid & 31) << 2;
  int s = edge_index[e * 2 + 0];
  int d = edge_index[e * 2 + 1];
  float w = ew[e];
  const float4 hv = *reinterpret_cast<const float4*>(h + (size_t)s * INT_F + c);
  float* base = agg + (size_t)d * INT_F + c;
  atomic_add_f32(base + 0, w * hv.x);
  atomic_add_f32(base + 1, w * hv.y);
  atomic_add_f32(base + 2, w * hv.z);
  atomic_add_f32(base + 3, w * hv.w);
}

// ---- fp32 WMMA GEMM: one wave computes one 16x16 tile of D -----------------
// D = A(MxK) * B(KxN) + bias  (MODE 0)
// D = Cres + leakyrelu( (diag(rdenom)*A) * B + bias )   in-place ok (MODE 1)
// A row-major lda=K, B row-major ldb=N, D row-major ldd=N. M % 16 == 0.
template <int K, int N, int MODE>
__global__ void wmma_gemm_kernel(const float* __restrict__ A,
                                 const float* __restrict__ B,
                                 const float* __restrict__ bias,
                                 const float* __restrict__ rdenom,
                                 const float* Cres, float* D) {
  const int lane = threadIdx.x & 31;
  const int wv   = threadIdx.x >> 5;     // wave -> N tile
  const int row0 = blockIdx.x * 16;
  const int col0 = wv * 16;
  const int half = lane >> 4;            // 0: K={0,1}, 1: K={2,3} per ISA A/B f32 layout
  const int l    = lane & 15;
  const int am   = row0 + l;             // A row owned by this lane
  const int kb   = half * 2;

  float ascale = (MODE == 1) ? rdenom[am] : 1.0f;
  const float* Arow = A + (size_t)am * K;

  v8f acc = {};
#pragma unroll
  for (int k = 0; k < K; k += 4) {
    v2f a, b;
    a.x = Arow[k + kb] * ascale;
    a.y = Arow[k + kb + 1] * ascale;
    b.x = B[(size_t)(k + kb) * N + col0 + l];
    b.y = B[(size_t)(k + kb + 1) * N + col0 + l];
    // 8 args: (neg_a, A, neg_b, B, c_mod, C, reuse_a, reuse_b)
    acc = __builtin_amdgcn_wmma_f32_16x16x4_f32(false, a, false, b,
                                                (short)0, acc, false, false);
  }

  const float bcol = bias[col0 + l];
#pragma unroll
  for (int r = 0; r < 8; ++r) {
    int row = row0 + r + half * 8;       // C/D layout: VGPR r -> M=r (+8 for hi lanes)
    size_t idx = (size_t)row * N + col0 + l;
    float z = acc[r] + bcol;
    if (MODE == 1) {
      z = Cres[idx] + (z >= 0.0f ? z : NEG_SLOPE * z);
    }
    D[idx] = z;
  }
}

extern "C" void kernel_launch(void* const* d_in, const int* in_sizes, int n_in,
                              void* d_out, int out_size, void* d_ws, size_t ws_size,
                              hipStream_t stream) {
  const float* node_features = (const float*)d_in[0];
  const int*   edge_index    = (const int*)d_in[1];
  const float* edge_weights  = (const float*)d_in[2];
  const float* W_in          = (const float*)d_in[3];
  const float* b_in          = (const float*)d_in[4];
  const float* W_layers      = (const float*)d_in[5];
  const float* b_layers      = (const float*)d_in[6];
  const float* W_out         = (const float*)d_in[7];
  const float* b_out         = (const float*)d_in[8];

  const int n_nodes = in_sizes[0] / IN_F;  // 50000 (multiple of 16)
  const int n_edges = in_sizes[1] / 2;     // 625000

  float* out = (float*)d_out;
  float* h   = out + (size_t)n_nodes * OUT_F;   // second tuple output doubles as h
  float* agg    = (float*)d_ws;                 // n_nodes*INT_F floats
  float* rdenom = agg + (size_t)n_nodes * INT_F;// n_nodes floats

  // degrees -> 1/max(deg,1)
  hipMemsetAsync(rdenom, 0, (size_t)n_nodes * sizeof(float), stream);
  count_edges_kernel<<<(n_edges + 255) / 256, 256, 0, stream>>>(edge_index, rdenom, n_edges);
  finalize_rdenom_kernel<<<(n_nodes + 255) / 256, 256, 0, stream>>>(rdenom, n_nodes);

  const int mtiles = (n_nodes + 15) / 16;  // 3125

  // h = X @ W_in + b_in
  wmma_gemm_kernel<IN_F, INT_F, 0><<<mtiles, 32 * (INT_F / 16), 0, stream>>>(
      node_features, W_in, b_in, nullptr, nullptr, h);

  for (int i = 0; i < GNN_DEPTH; ++i) {
    hipMemsetAsync(agg, 0, (size_t)n_nodes * INT_F * sizeof(float), stream);
    scatter_kernel<<<(n_edges * 32 + 255) / 256, 256, 0, stream>>>(
        h, edge_index, edge_weights, agg, n_edges);
    // h = h + leakyrelu((agg/denom) @ W_i + b_i)
    wmma_gemm_kernel<INT_F, INT_F, 1><<<mtiles, 32 * (INT_F / 16), 0, stream>>>(
        agg, W_layers + (size_t)i * INT_F * INT_F, b_layers + (size_t)i * INT_F,
        rdenom, h, h);
  }

  // out = h @ W_out + b_out
  wmma_gemm_kernel<INT_F, OUT_F, 0><<<mtiles, 32 * (OUT_F / 16), 0, stream>>>(
      h, W_out, b_out, nullptr, nullptr, out);
}